// CTRNN_MD_67396626809428
// MI455X (gfx1250) — compile-verified
//
#include <hip/hip_runtime.h>
#include <hip/hip_bf16.h>

// ---------------------------------------------------------------------------
// CTRNN + MD gating scan for MI455X (gfx1250).
//
// Structure:
//   1) init_kernel      : reset plastic weights / traces / barrier counter
//   2) wmma_xwT (x2)    : ext_all  = X @ W_in2h^T + b  -> d_out (reused as ext)
//                         ectx_all = X @ W_in2ctx^T    -> ws
//      (f32 WMMA 16x16x4, one wave per 16x16 tile)
//   3) scan_kernel      : 16 persistent blocks, 2000 sequential steps,
//                         global arrive/spin barrier between phases.
// ---------------------------------------------------------------------------

#define Tn     2000
#define INn    1024
#define Hn     2048
#define HCTXn  4096
#define SUBn   512
#define MDn    128
#define MDACT  16
#define ALPHA  0.1f
#define SUBPROB 0.5f
#define NOISESTD 0.01f
#define MDRATE 0.0125f            // DT_MD / (TAU_MD) = 0.001 / 0.08
#define TRRATE 0.001f             // 1 / TAU_TRACE
#define HEBBLR 1e-4f
#define PREPART 3276              // int(0.8 * 4096)

#define Gb 16                     // persistent blocks in scan
#define SCANT 256                 // threads per scan block

typedef __attribute__((ext_vector_type(2))) float v2f;
typedef __attribute__((ext_vector_type(8))) float v8f;

// ------------------------------- GEMM (WMMA) -------------------------------
// C[M,N] = X[M,K] * W[N,K]^T (+ bias[N]).  One wave per 16x16 tile.
// A (16x4 f32): lanes 0-15 hold K={k,k+1} for M=lane; lanes 16-31 K={k+2,k+3}.
// B (4x16 f32): lanes 0-15 hold K={k,k+1} for N=lane; lanes 16-31 K={k+2,k+3}.
// C/D: VGPR v -> M = v (+8 for upper lane half), N = lane & 15.
__global__ __launch_bounds__(256) void wmma_xwT(
    const float* __restrict__ X, const float* __restrict__ W,
    const float* __restrict__ bias, float* __restrict__ C,
    int M, int N, int K)
{
  const int lane  = threadIdx.x & 31;
  const int wave  = (blockIdx.x * blockDim.x + threadIdx.x) >> 5;
  const int tilesN = N >> 4;
  const int tm = wave / tilesN;
  const int tn = wave - tm * tilesN;
  if (tm * 16 >= M) return;                       // wave-uniform; EXEC stays full

  const int mrow  = tm * 16 + (lane & 15);        // A row for this lane
  const int ncol  = tn * 16 + (lane & 15);        // B/W row (= output column)
  const int khalf = (lane >> 4) << 1;             // 0 or 2

  const float* __restrict__ Xp = X + (size_t)mrow * K + khalf;
  const float* __restrict__ Wp = W + (size_t)ncol * K + khalf;

  v8f acc = {};
  for (int k = 0; k < K; k += 4) {
    v2f a = *(const v2f*)(Xp + k);
    v2f b = *(const v2f*)(Wp + k);
    acc = __builtin_amdgcn_wmma_f32_16x16x4_f32(
        /*neg_a=*/false, a, /*neg_b=*/false, b,
        /*c_mod=*/(short)0, acc, /*reuse_a=*/false, /*reuse_b=*/false);
  }

  const float bv = bias ? bias[ncol] : 0.0f;
  const int mbase = tm * 16 + ((lane >> 4) << 3);  // +8 for upper half
  #pragma unroll
  for (int v = 0; v < 8; ++v)
    C[(size_t)(mbase + v) * N + ncol] = acc[v] + bv;
}

// ------------------------------- helpers -----------------------------------
__device__ __forceinline__ float wave_sum(float v) {
  #pragma unroll
  for (int o = 16; o > 0; o >>= 1) v += __shfl_xor(v, o, 32);
  return v;
}

__device__ __forceinline__ unsigned f2ord(float f) {
  unsigned u = __float_as_uint(f);
  return (u & 0x80000000u) ? ~u : (u | 0x80000000u);
}
__device__ __forceinline__ float ord2f(unsigned u) {
  unsigned v = (u & 0x80000000u) ? (u & 0x7FFFFFFFu) : ~u;
  return __uint_as_float(v);
}

// Global arrive/spin barrier across the Gb persistent blocks (all co-resident:
// 16 blocks x 256 thr, ~20KB LDS each -- trivially fits the WGP array).
__device__ __forceinline__ void gbar(unsigned* cnt, unsigned expected) {
  __syncthreads();
  if (threadIdx.x == 0) {
    __threadfence();
    atomicAdd(cnt, 1u);
    while (__hip_atomic_load(cnt, __ATOMIC_ACQUIRE, __HIP_MEMORY_SCOPE_AGENT) < expected)
      __builtin_amdgcn_s_sleep(2);
    __threadfence();
  }
  __syncthreads();
}

// ------------------------------- init --------------------------------------
__global__ void init_kernel(const float* __restrict__ hidden0,
                            const float* __restrict__ wPFC2MD,
                            float* h, float* pre, float* post, float* md_inp,
                            float* w_p2m, unsigned* barcnt)
{
  const int i = blockIdx.x * blockDim.x + threadIdx.x;
  const int stride = gridDim.x * blockDim.x;
  if (i == 0) *barcnt = 0u;
  if (i < Hn)    h[i] = hidden0[i];
  if (i < HCTXn) pre[i] = 0.0f;
  if (i < MDn)   { post[i] = 0.0f; md_inp[i] = 0.0f; }
  for (int e = i; e < MDn * HCTXn; e += stride) w_p2m[e] = wPFC2MD[e];
}

// ------------------------------- scan --------------------------------------
__global__ __launch_bounds__(SCANT) void scan_kernel(
    const float* __restrict__ noise,       // [T,HCTX]
    const float* __restrict__ mask_rand,   // [T,SUB]
    const float* __restrict__ w_h2h,       // [H,H]
    const float* __restrict__ b_h2h,       // [H]
    const float* __restrict__ wMD2PFC,     // [H,MD]
    const float* __restrict__ wMD2PFCMult, // [H,MD]
    const int*   __restrict__ sub_id_p,
    const float* __restrict__ ectx,        // [T,HCTX] precomputed
    float* __restrict__ hs,                // d_out [T,H]: holds ext, overwritten by h
    float* h, float* r, float* pre, float* rec,
    float* md_inp, float* post, float* md_out, float* out_tr,
    float* md2pfc, float* md2pfcm, float* thr,
    float* w_p2m, unsigned* barcnt)
{
  const int tid   = threadIdx.x;
  const int gid   = blockIdx.x * SCANT + tid;   // 0..4095
  const int lane  = tid & 31;
  const int gwave = gid >> 5;                   // 0..127
  const int lo    = sub_id_p[0] * SUBn;
  unsigned gen = 0;

  __shared__ float    sbuf[HCTXn];  // block0: r stage; other blocks: h stage
  __shared__ unsigned hist[256];
  __shared__ float    sv[MDn];
  __shared__ float    smk[MDn];
  __shared__ float    rsum[SCANT];
  __shared__ int      rcnt[SCANT];
  __shared__ unsigned spref;
  __shared__ int      srank;

  for (int t = 0; t < Tn; ++t) {
    // ---- P0: pfc_ctx activity r and pre-trace (4096 elems, 1/thread) ------
    {
      const int j = gid;
      float m = 0.0f;
      const int js = j - lo;
      if (js >= 0 && js < SUBn)
        m = (mask_rand[(size_t)t * SUBn + js] < SUBPROB) ? 1.0f : 0.0f;
      float rv = ectx[(size_t)t * HCTXn + j] * m
               + NOISESTD * noise[(size_t)t * HCTXn + j];
      rv = fmaxf(rv, 0.0f);
      r[j] = rv;
      const float pv = pre[j];
      pre[j] = pv + TRRATE * (rv - pv);
    }
    gbar(barcnt, ++gen * Gb);

    // ---- P1: block0 -> MD matvec then serial P2; blocks1..15 -> rec -------
    if (blockIdx.x == 0) {
      for (int j = tid; j < HCTXn; j += SCANT) sbuf[j] = r[j];
      __syncthreads();
      // md_inp += rate * (-md_inp + w_p2m @ r): 8 waves x 16 rows
      const int wv = tid >> 5;
      for (int rr = wv; rr < MDn; rr += 8) {
        float acc = 0.0f;
        const float* wp = w_p2m + (size_t)rr * HCTXn;
        for (int c = lane; c < HCTXn; c += 32) acc = fmaf(wp[c], sbuf[c], acc);
        acc = wave_sum(acc);
        if (lane == 0) {
          float mi = md_inp[rr];
          mi = mi + MDRATE * (acc - mi);
          md_inp[rr] = mi;
          sv[rr] = mi;
        }
      }
      __syncthreads();

      // -- P2 (serial, block0 only) --
      // WTA(md_inp) -> md_out  (strict '>' => lowest index wins ties, as top_k)
      if (tid < MDn) smk[tid] = 0.0f;
      __syncthreads();
      if (tid == 0) {
        for (int it = 0; it < MDACT; ++it) {
          float best = -3.4e38f; int bi = 0;
          for (int i2 = 0; i2 < MDn; ++i2)
            if (sv[i2] > best) { best = sv[i2]; bi = i2; }
          sv[bi] = -3.4e38f; smk[bi] = 1.0f;
        }
      }
      __syncthreads();
      if (tid < MDn) {
        md_out[tid] = smk[tid];
        float po = post[tid] + TRRATE * (smk[tid] - post[tid]);
        post[tid] = po;
        sv[tid] = po;           // reuse for WTA(post)
        smk[tid] = 0.0f;
      }
      __syncthreads();
      if (tid == 0) {
        for (int it = 0; it < MDACT; ++it) {
          float best = -3.4e38f; int bi = 0;
          for (int i2 = 0; i2 < MDn; ++i2)
            if (sv[i2] > best) { best = sv[i2]; bi = i2; }
          sv[bi] = -3.4e38f; smk[bi] = 1.0f;
        }
      }
      __syncthreads();
      if (tid < MDn) out_tr[tid] = smk[tid];

      // Exact k-th order statistic of pre[4096] via radix select on
      // float-ordered uints, then thr = mean of lowest PREPART values.
      unsigned prefix = 0; int rank = PREPART - 1;
      for (int pass = 3; pass >= 0; --pass) {
        for (int b = tid; b < 256; b += SCANT) hist[b] = 0u;
        __syncthreads();
        const unsigned pm = (pass == 3) ? 0u : (0xFFFFFFFFu << ((pass + 1) * 8));
        for (int j = tid; j < HCTXn; j += SCANT) {
          const unsigned u = f2ord(pre[j]);
          if ((u & pm) == (prefix & pm))
            atomicAdd(&hist[(u >> (pass * 8)) & 255u], 1u);
        }
        __syncthreads();
        if (tid == 0) {
          int racc = rank; unsigned bsel = 255u;
          for (unsigned b = 0; b < 256u; ++b) {
            const unsigned c = hist[b];
            if (racc < (int)c) { bsel = b; break; }
            racc -= (int)c;
          }
          spref = prefix | (bsel << (pass * 8));
          srank = racc;
        }
        __syncthreads();
        prefix = spref; rank = srank;
      }
      const float vk = ord2f(prefix);
      float lsum = 0.0f; int lcnt = 0;
      for (int j = tid; j < HCTXn; j += SCANT) {
        const float p = pre[j];
        if (p < vk) { lsum += p; lcnt++; }
      }
      rsum[tid] = lsum; rcnt[tid] = lcnt;
      __syncthreads();
      for (int s = SCANT / 2; s > 0; s >>= 1) {
        if (tid < s) { rsum[tid] += rsum[tid + s]; rcnt[tid] += rcnt[tid + s]; }
        __syncthreads();
      }
      if (tid == 0)
        thr[0] = (rsum[0] + (float)(PREPART - rcnt[0]) * vk) / (float)PREPART;
    } else {
      // rec = h @ w_h2h^T + b_h2h : 120 waves, ~17 rows each, wave-coop dot
      for (int j = tid; j < Hn; j += SCANT) sbuf[j] = h[j];
      __syncthreads();
      const int wv = gwave - 8;            // 0..119
      for (int rr = wv; rr < Hn; rr += 120) {
        float acc = 0.0f;
        const float* wp = w_h2h + (size_t)rr * Hn;
        __builtin_prefetch(wp + Hn, 0, 1); // global_prefetch_b8 toward next row
        for (int c = lane; c < Hn; c += 32) acc = fmaf(wp[c], sbuf[c], acc);
        acc = wave_sum(acc);
        if (lane == 0) rec[rr] = acc + b_h2h[rr];
      }
    }
    gbar(barcnt, ++gen * Gb);

    // ---- P3: MD->PFC gates + Hebbian plasticity ---------------------------
    for (int rr = gwave; rr < Hn; rr += 128) {
      float a0 = 0.0f, a1 = 0.0f;
      const float* wa = wMD2PFC     + (size_t)rr * MDn;
      const float* wm = wMD2PFCMult + (size_t)rr * MDn;
      for (int c = lane; c < MDn; c += 32) {
        const float mo = md_out[c];
        a0 = fmaf(wa[c], mo, a0);
        a1 = fmaf(wm[c], mo, a1);
      }
      a0 = wave_sum(a0); a1 = wave_sum(a1);
      if (lane == 0) { md2pfc[rr] = a0 * (1.0f / (float)MDn); md2pfcm[rr] = a1; }
    }
    {
      // dmask = 0.05*exp(log(20)*o*p) with binary o,p  ==  (o&&p) ? 1 : 0.05
      const float th = thr[0];
      for (int e = gid; e < MDn * HCTXn; e += Gb * SCANT) {
        const int mrow = e >> 12;
        const int j    = e & (HCTXn - 1);
        const float o = out_tr[mrow];
        const float p = (pre[j] > th) ? 1.0f : 0.0f;
        const float delta = 0.5f * HEBBLR * (o - 0.5f) * (p - 0.5f);
        const float dm = (o * p > 0.5f) ? 1.0f : 0.05f;
        float w = w_p2m[e] + delta * dm;
        w_p2m[e] = fminf(fmaxf(w, 0.0f), 1.0f);
      }
    }
    gbar(barcnt, ++gen * Gb);

    // ---- P4: hidden update; overwrite ext row of d_out with h[t] ----------
    if (gid < Hn) {
      const int i = gid;
      const float ext = hs[(size_t)t * Hn + i];
      const float pa  = ext + rec[i] * md2pfcm[i] + md2pfc[i];
      float hn = h[i] * (1.0f - ALPHA) + ALPHA * pa;
      hn = fmaxf(hn, 0.0f);
      h[i] = hn;
      hs[(size_t)t * Hn + i] = hn;
    }
    gbar(barcnt, ++gen * Gb);
  }
}

// ------------------------------- launch ------------------------------------
extern "C" void kernel_launch(void* const* d_in, const int* in_sizes, int n_in,
                              void* d_out, int out_size, void* d_ws, size_t ws_size,
                              hipStream_t stream) {
  const float* x        = (const float*)d_in[0];   // [T,1,IN]
  const float* hidden0  = (const float*)d_in[1];   // [1,H]
  const float* noise    = (const float*)d_in[2];   // [T,1,HCTX]
  const float* mask_rnd = (const float*)d_in[3];   // [T,SUB]
  const float* w_in2h   = (const float*)d_in[4];   // [H,IN]
  const float* b_in2h   = (const float*)d_in[5];   // [H]
  const float* w_h2h    = (const float*)d_in[6];   // [H,H]
  const float* b_h2h    = (const float*)d_in[7];   // [H]
  const float* w_in2ctx = (const float*)d_in[8];   // [HCTX,IN]
  const float* wPFC2MD  = (const float*)d_in[9];   // [MD,HCTX]
  const float* wMD2PFC  = (const float*)d_in[10];  // [H,MD]
  const float* wMD2PFCm = (const float*)d_in[11];  // [H,MD]
  const int*   sub_id   = (const int*)d_in[12];    // [1]
  float* hs = (float*)d_out;                       // [T,H]

  // workspace layout (floats)
  float* ws = (float*)d_ws;
  size_t o = 0;
  float* ectx    = ws + o; o += (size_t)Tn * HCTXn;   // 8,192,000
  float* w_p2m   = ws + o; o += (size_t)MDn * HCTXn;  // 524,288
  float* h       = ws + o; o += Hn;
  float* r       = ws + o; o += HCTXn;
  float* pre     = ws + o; o += HCTXn;
  float* rec     = ws + o; o += Hn;
  float* md_inp  = ws + o; o += MDn;
  float* post    = ws + o; o += MDn;
  float* md_out  = ws + o; o += MDn;
  float* out_tr  = ws + o; o += MDn;
  float* md2pfc  = ws + o; o += Hn;
  float* md2pfcm = ws + o; o += Hn;
  float* thr     = ws + o; o += 4;
  unsigned* barcnt = (unsigned*)(ws + o);

  // 1) init plastic weights / traces / barrier
  init_kernel<<<1024, 256, 0, stream>>>(hidden0, wPFC2MD,
                                        h, pre, post, md_inp, w_p2m, barcnt);

  // 2) WMMA input projections (f32 16x16x4); ext goes straight into d_out
  {
    const int waves1 = (Tn / 16) * (Hn / 16);      // 16000 -> 2000 blocks
    wmma_xwT<<<waves1 / 8, 256, 0, stream>>>(x, w_in2h, b_in2h, hs, Tn, Hn, INn);
    const int waves2 = (Tn / 16) * (HCTXn / 16);   // 32000 -> 4000 blocks
    wmma_xwT<<<waves2 / 8, 256, 0, stream>>>(x, w_in2ctx, nullptr, ectx, Tn, HCTXn, INn);
  }

  // 3) sequential scan: 16 persistent blocks, global barrier between phases
  scan_kernel<<<Gb, SCANT, 0, stream>>>(noise, mask_rnd, w_h2h, b_h2h,
                                        wMD2PFC, wMD2PFCm, sub_id,
                                        ectx, hs,
                                        h, r, pre, rec, md_inp, post,
                                        md_out, out_tr, md2pfc, md2pfcm, thr,
                                        w_p2m, barcnt);
}